// Net_26852135534930
// MI455X (gfx1250) — compile-verified
//
#include <hip/hip_runtime.h>
#include <hip/hip_bf16.h>

#define F_IN     256
#define HID      16
#define NCLS     10

typedef __attribute__((ext_vector_type(2))) float v2f;
typedef __attribute__((ext_vector_type(8))) float v8f;

// ---------------- degree / normalization ----------------

__global__ void k_deg_init(float* __restrict__ deg, int n) {
    int i = blockIdx.x * blockDim.x + threadIdx.x;
    if (i < n) deg[i] = 1.0f;                 // self-loop
}

__global__ void k_deg_count(const int* __restrict__ col, float* __restrict__ deg, int e) {
    int i = blockIdx.x * blockDim.x + threadIdx.x;
    if (i < e) unsafeAtomicAdd(&deg[col[i]], 1.0f);
}

__global__ void k_deg_rsqrt(float* __restrict__ deg, int n) {
    int i = blockIdx.x * blockDim.x + threadIdx.x;
    if (i < n) deg[i] = rsqrtf(deg[i]);       // deg >= 1 always (self-loop)
}

// ---------------- GEMM1: bufA = x @ W1  (N x 256 @ 256 x 16) ----------------
// One wave per 16-row output tile. fp32 WMMA 16x16x4, 64 K-steps.
// A layout (16x4 f32): lane m=lane&15 holds row m; kg=lane>>4 selects K pair
//   {2kg, 2kg+1} in VGPR0/1 -> contiguous float2 load from x.
// B layout (4x16): VGPR g, lane -> K = g + 2*kg, N = m.
// C/D layout: VGPR g -> row M = g + 8*kg, col N = m.

__global__ void k_gemm1(const float* __restrict__ x, const float* __restrict__ W,
                        float* __restrict__ out, int ntiles) {
    const int wave = (blockIdx.x * blockDim.x + threadIdx.x) >> 5;
    if (wave >= ntiles) return;
    const int lane = threadIdx.x & 31;
    const int m  = lane & 15;
    const int kg = lane >> 4;
    const long rowbase = (long)(wave * 16 + m) * F_IN;
    v8f acc = {};
    for (int k0 = 0; k0 < F_IN; k0 += 4) {
        const int ka = k0 + kg * 2;
        v2f a;
        a.x = x[rowbase + ka];
        a.y = x[rowbase + ka + 1];
        v2f b;
        b.x = W[(long)ka * HID + m];
        b.y = W[(long)(ka + 1) * HID + m];
        acc = __builtin_amdgcn_wmma_f32_16x16x4_f32(false, a, false, b,
                                                    (short)0, acc, false, false);
    }
    const long obase = (long)wave * 16 * HID;
    #pragma unroll
    for (int g = 0; g < 8; ++g)
        out[obase + (long)(g + kg * 8) * HID + m] = acc[g];
}

// ---------------- aggregation ----------------
// init: out[v] = dis[v]^2 * h[v]   (the row==col self-loop term)

__global__ void k_agg_init(const float* __restrict__ h, const float* __restrict__ dis,
                           float* __restrict__ out, int n) {
    int t = blockIdx.x * blockDim.x + threadIdx.x;
    if (t >= n * HID) return;
    int v = t >> 4;
    float s = dis[v];
    out[t] = s * s * h[t];
}

// edges: out[v][f] += dis[u]*dis[v]*h[u][f]; one lane per (edge, feature)

__global__ void k_agg_edge(const int* __restrict__ row, const int* __restrict__ col,
                           const float* __restrict__ dis, const float* __restrict__ h,
                           float* __restrict__ out, int e, int nf) {
    long t = (long)blockIdx.x * blockDim.x + threadIdx.x;
    long ei = t >> 4;
    int  f  = (int)(t & 15);
    if (ei >= e || f >= nf) return;
    int u = row[ei], v = col[ei];
    float w = dis[u] * dis[v];
    unsafeAtomicAdd(&out[(long)v * HID + f], w * h[(long)u * HID + f]);
}

// ---------------- GEMM2: bufA = relu(bufB + b1) @ pad16(W2) ----------------

__global__ void k_gemm2(const float* __restrict__ hin, const float* __restrict__ b1,
                        const float* __restrict__ W2, float* __restrict__ out, int ntiles) {
    const int wave = (blockIdx.x * blockDim.x + threadIdx.x) >> 5;
    if (wave >= ntiles) return;
    const int lane = threadIdx.x & 31;
    const int m  = lane & 15;
    const int kg = lane >> 4;
    const long rowbase = (long)(wave * 16 + m) * HID;
    v8f acc = {};
    #pragma unroll
    for (int k0 = 0; k0 < HID; k0 += 4) {
        const int ka = k0 + kg * 2;
        float a0 = hin[rowbase + ka]     + b1[ka];
        float a1 = hin[rowbase + ka + 1] + b1[ka + 1];
        v2f a;
        a.x = a0 > 0.0f ? a0 : 0.0f;      // relu fused into the load
        a.y = a1 > 0.0f ? a1 : 0.0f;
        v2f b;
        b.x = (m < NCLS) ? W2[ka * NCLS + m]       : 0.0f;
        b.y = (m < NCLS) ? W2[(ka + 1) * NCLS + m] : 0.0f;
        acc = __builtin_amdgcn_wmma_f32_16x16x4_f32(false, a, false, b,
                                                    (short)0, acc, false, false);
    }
    const long obase = (long)wave * 16 * HID;
    #pragma unroll
    for (int g = 0; g < 8; ++g)
        out[obase + (long)(g + kg * 8) * HID + m] = acc[g];
}

// ---------------- log_softmax over 10 classes ----------------

__global__ void k_logsoftmax(const float* __restrict__ agg, const float* __restrict__ b2,
                             float* __restrict__ out, int n) {
    int v = blockIdx.x * blockDim.x + threadIdx.x;
    if (v >= n) return;
    float vals[NCLS];
    float mx = -3.402823466e38f;
    #pragma unroll
    for (int c = 0; c < NCLS; ++c) {
        vals[c] = agg[(long)v * HID + c] + b2[c];
        mx = fmaxf(mx, vals[c]);
    }
    float s = 0.0f;
    #pragma unroll
    for (int c = 0; c < NCLS; ++c) s += __expf(vals[c] - mx);
    float lse = mx + __logf(s);
    #pragma unroll
    for (int c = 0; c < NCLS; ++c) out[(long)v * NCLS + c] = vals[c] - lse;
}

// ---------------- launch ----------------

extern "C" void kernel_launch(void* const* d_in, const int* in_sizes, int n_in,
                              void* d_out, int out_size, void* d_ws, size_t ws_size,
                              hipStream_t stream) {
    const float* x  = (const float*)d_in[0];
    const int*   ei = (const int*)d_in[1];   // [2, E]
    const float* W1 = (const float*)d_in[2];
    const float* b1 = (const float*)d_in[3];
    const float* W2 = (const float*)d_in[4];
    const float* b2 = (const float*)d_in[5];

    const int N = in_sizes[0] / F_IN;
    const int E = in_sizes[1] / 2;
    const int* row = ei;
    const int* col = ei + E;

    // workspace: dis[N] | bufA[N*16] | bufB[N*16]  (~13.3 MB)
    float* dis  = (float*)d_ws;
    float* bufA = dis + (((size_t)N + 127) & ~(size_t)127);
    float* bufB = bufA + (size_t)N * HID;

    const int ntiles = (N + 15) / 16;
    const int TB = 256;

    // 1. symmetric degree normalization
    k_deg_init <<<(N + TB - 1) / TB, TB, 0, stream>>>(dis, N);
    k_deg_count<<<(E + TB - 1) / TB, TB, 0, stream>>>(col, dis, E);
    k_deg_rsqrt<<<(N + TB - 1) / TB, TB, 0, stream>>>(dis, N);

    // 2. layer-1 dense: bufA = x @ W1
    k_gemm1<<<(ntiles + 3) / 4, 128, 0, stream>>>(x, W1, bufA, ntiles);

    // 3. layer-1 aggregation into bufB
    k_agg_init<<<((size_t)N * HID + TB - 1) / TB, TB, 0, stream>>>(bufA, dis, bufB, N);
    {
        long nt = (long)E * HID;
        k_agg_edge<<<(nt + TB - 1) / TB, TB, 0, stream>>>(row, col, dis, bufA, bufB, E, HID);
    }

    // 4. layer-2 dense (relu + b1 fused): bufA = relu(bufB+b1) @ pad(W2)
    k_gemm2<<<(ntiles + 3) / 4, 128, 0, stream>>>(bufB, b1, W2, bufA, ntiles);

    // 5. layer-2 aggregation into bufB (10 live features)
    k_agg_init<<<((size_t)N * HID + TB - 1) / TB, TB, 0, stream>>>(bufA, dis, bufB, N);
    {
        long nt = (long)E * HID;
        k_agg_edge<<<(nt + TB - 1) / TB, TB, 0, stream>>>(row, col, dis, bufA, bufB, E, NCLS);
    }

    // 6. +b2, log_softmax -> d_out [N,10]
    k_logsoftmax<<<(N + TB - 1) / TB, TB, 0, stream>>>(bufB, b2, (float*)d_out, N);
}